// VectorField_14929306321120
// MI455X (gfx1250) — compile-verified
//
#include <hip/hip_runtime.h>

// VectorField.set_from_points for MI455X (gfx1250):
//   phase 0: zero accumulators (memset nodes, graph-capturable)
//   phase 1: scatter — per-point global_atomic_add_f32 x4 into an accumulator
//            that fits in the 192MB L2 (113MB working set); streamed inputs use
//            non-temporal (th:NT) loads so they do not evict the accumulator.
//   phase 2: finalize — out = sum / (weight + eps), in place over d_out.
//
// No WMMA: the reference contains no matmul; this is a bandwidth/atomic-bound
// scatter. HBM roofline ~300MB -> ~13us @ 23.3TB/s.

namespace {
constexpr int   RES_X = 192, RES_Y = 192, RES_Z = 192;
constexpr int   NCELL = RES_X * RES_Y * RES_Z;   // 7,077,888
constexpr float EPSF  = 1e-8f;
}

__device__ __forceinline__ void atomic_add_agent_f32(float* p, float v) {
    // Relaxed, device(agent)-scope, result discarded -> non-returning
    // global_atomic_add_f32 (STOREcnt path), resolved in L2.
    (void)__hip_atomic_fetch_add(p, v, __ATOMIC_RELAXED, __HIP_MEMORY_SCOPE_AGENT);
}

__global__ __launch_bounds__(256) void vf_scatter(
    const float* __restrict__ points,
    const float* __restrict__ values,
    float* __restrict__ sums,    // [NCELL*3] accumulator == d_out
    float* __restrict__ wts,     // [NCELL]   weight accumulator (scratch)
    int n)
{
    const int i = blockIdx.x * 256 + threadIdx.x;
    if (i >= n) return;

    // Reference: coords = (p - 0) / (1 + 1e-8) * 192. In fp32, 1.0f+1e-8f == 1.0f,
    // so the transform is exactly p * 192.0f.
    const float scale = 192.0f / (1.0f + 1e-8f);   // == 192.0f in fp32

    const float* __restrict__ p = points + 3ull * (unsigned)i;
    const float cx = __builtin_nontemporal_load(p + 0) * scale;
    const float cy = __builtin_nontemporal_load(p + 1) * scale;
    const float cz = __builtin_nontemporal_load(p + 2) * scale;

    // trunc-toward-zero cast == .astype(int32) for these non-negative coords,
    // then clamp to [0, res-1] (matches jnp.clip after cast).
    int ix = (int)cx; ix = ix < 0 ? 0 : (ix > RES_X - 1 ? RES_X - 1 : ix);
    int iy = (int)cy; iy = iy < 0 ? 0 : (iy > RES_Y - 1 ? RES_Y - 1 : iy);
    int iz = (int)cz; iz = iz < 0 ? 0 : (iz > RES_Z - 1 ? RES_Z - 1 : iz);

    const int flat = (ix * RES_Y + iy) * RES_Z + iz;

    const float* __restrict__ v = values + 3ull * (unsigned)i;
    const float vx = __builtin_nontemporal_load(v + 0);
    const float vy = __builtin_nontemporal_load(v + 1);
    const float vz = __builtin_nontemporal_load(v + 2);

    float* __restrict__ s = sums + 3ull * (unsigned)flat;
    atomic_add_agent_f32(s + 0, vx);
    atomic_add_agent_f32(s + 1, vy);
    atomic_add_agent_f32(s + 2, vz);
    atomic_add_agent_f32(wts + flat, 1.0f);
}

__global__ __launch_bounds__(256) void vf_finalize(
    float* __restrict__ sums,          // in/out, [NCELL*3]
    const float* __restrict__ wts)     // [NCELL]
{
    const int i = blockIdx.x * 256 + threadIdx.x;
    if (i >= NCELL) return;

    const float w = wts[i] + EPSF;     // reads mostly hit L2 (accumulator resident)
    float* __restrict__ s = sums + 3ull * (unsigned)i;
    // True IEEE division to match reference rounding of grid/(w+eps).
    s[0] = s[0] / w;
    s[1] = s[1] / w;
    s[2] = s[2] / w;
}

extern "C" void kernel_launch(void* const* d_in, const int* in_sizes, int n_in,
                              void* d_out, int out_size, void* d_ws, size_t ws_size,
                              hipStream_t stream)
{
    (void)n_in; (void)out_size; (void)ws_size;

    const float* points = (const float*)d_in[0];   // (N,3) f32
    const float* values = (const float*)d_in[1];   // (N,3) f32
    float* sums = (float*)d_out;                   // (192,192,192,3) f32
    float* wts  = (float*)d_ws;                    // NCELL f32 scratch (28.3 MB)

    const int n = in_sizes[0] / 3;                 // flat count -> point count

    // Zero accumulators every call (harness poisons d_out/d_ws; must be
    // deterministic). Memset nodes are graph-capturable.
    hipMemsetAsync(sums, 0, (size_t)NCELL * 3 * sizeof(float), stream);
    hipMemsetAsync(wts,  0, (size_t)NCELL * sizeof(float), stream);

    const int blocks = (n + 255) / 256;
    vf_scatter<<<blocks, 256, 0, stream>>>(points, values, sums, wts, n);

    const int fblocks = (NCELL + 255) / 256;
    vf_finalize<<<fblocks, 256, 0, stream>>>(sums, wts);
}